// ChamferLoss_11742440587475
// MI455X (gfx1250) — compile-verified
//
#include <hip/hip_runtime.h>

typedef float v2f __attribute__((ext_vector_type(2)));
typedef float v8f __attribute__((ext_vector_type(8)));

#define WAVES_PER_BLOCK 8
#define BLOCK_THREADS   256
#define ROWTILES        2      // row tiles (16 rows each) per wave

__global__ void chamfer_zero_kernel(float* out) { out[0] = 0.0f; }

// Each wave owns ROWTILES consecutive 16-row tiles of P (within one batch) and
// scans all 16-col tiles of Q.
//
// WMMA trick: K=4 with D=3 leaves one K slot free.
//   A[m,0..2] = -2 * p[m],  A[m,3] = 1
//   B[0..2,n] =  q[n],      B[3,n] = ||q_n||^2
//   C[v]      = ||p_row||^2           (loop-invariant!)
// => D[m][n] = ||p_m||^2 + ||q_n||^2 - 2<p_m,q_n>  directly from one WMMA.
//
// D-matrix layout (16x16 f32, 8 VGPRs, wave32):
//   lane l, reg v  ->  row m = v + 8*(l>=16), col n = l%16
__global__ __launch_bounds__(BLOCK_THREADS)
void chamfer_dir_kernel(const float* __restrict__ P,   // [B, Np, 3] query points
                        const float* __restrict__ Q,   // [B, Nq, 3] target points
                        int Np, int Nq, int numPairs, float scale,
                        float* __restrict__ out)
{
    const int lane = threadIdx.x & 31;
    const int wave = threadIdx.x >> 5;
    const int pair = blockIdx.x * WAVES_PER_BLOCK + wave;
    if (pair >= numPairs) return;            // wave-uniform exit: EXEC stays all-ones

    const int pairsPerBatch = Np >> (4 + 1); // (Np/16)/ROWTILES, ROWTILES==2
    const int b   = pair / pairsPerBatch;
    const int rt0 = (pair - b * pairsPerBatch) * ROWTILES;

    const float* Pb = P + (size_t)b * (size_t)Np * 3u;
    const float* Qb = Q + (size_t)b * (size_t)Nq * 3u;

    const int kh = lane >> 4;   // K-half: lanes 0-15 -> K={0,1}, lanes 16-31 -> K={2,3}
    const int li = lane & 15;

    v2f A[ROWTILES];
    v8f C[ROWTILES];
    #pragma unroll
    for (int r = 0; r < ROWTILES; ++r) {
        const int   prow = (rt0 + r) * 16 + li;
        const float p0 = Pb[prow * 3 + 0];
        const float p1 = Pb[prow * 3 + 1];
        const float p2 = Pb[prow * 3 + 2];
        // A tile: coordinates scaled by -2; K=3 slot holds 1.0 (picks up ||q||^2 from B).
        A[r].x = kh ? (p2 * -2.0f) : (p0 * -2.0f);
        A[r].y = kh ? 1.0f         : (p1 * -2.0f);
        // ||p||^2 of row (rt*16 + li) lives on lane li of both halves; redistribute
        // to the D layout (reg v needs row v + 8*kh) -> loop-invariant accumulator.
        const float pn = p0 * p0 + p1 * p1 + p2 * p2;
        #pragma unroll
        for (int v = 0; v < 8; ++v)
            C[r][v] = __shfl(pn, v + 8 * kh, 32);
    }

    float rmin[ROWTILES][8];
    #pragma unroll
    for (int r = 0; r < ROWTILES; ++r)
        #pragma unroll
        for (int v = 0; v < 8; ++v) rmin[r][v] = 3.402823466e38f;

    // ---- Stream all column tiles of Q (L2-resident working set).
    #pragma unroll 2
    for (int ct = 0; ct < Nq; ct += 16) {
        const int   qrow = ct + li;
        const float q0 = Qb[qrow * 3 + 0];
        const float q1 = Qb[qrow * 3 + 1];
        const float q2 = Qb[qrow * 3 + 2];
        const float qn = q0 * q0 + q1 * q1 + q2 * q2;

        // B tile: col n = li;  K={0,1} -> (q0,q1),  K={2,3} -> (q2, ||q||^2).
        v2f Bv;
        Bv.x = kh ? q2 : q0;
        Bv.y = kh ? qn : q1;

        #pragma unroll
        for (int r = 0; r < ROWTILES; ++r) {
            v8f d = __builtin_amdgcn_wmma_f32_16x16x4_f32(
                /*neg_a=*/false, A[r], /*neg_b=*/false, Bv,
                /*c_mod=*/(short)0, C[r], /*reuse_a=*/false, /*reuse_b=*/false);
            #pragma unroll
            for (int v = 0; v < 8; ++v)
                rmin[r][v] = fminf(rmin[r][v], d[v]);
        }
    }

    // ---- Reduce over the 16 columns within each lane half (butterfly),
    //      then sum the row-mins of all row tiles this wave owns.
    float rowsum = 0.0f;
    #pragma unroll
    for (int r = 0; r < ROWTILES; ++r) {
        #pragma unroll
        for (int m = 1; m <= 8; m <<= 1) {
            #pragma unroll
            for (int v = 0; v < 8; ++v)
                rmin[r][v] = fminf(rmin[r][v], __shfl_xor(rmin[r][v], m, 32));
        }
        #pragma unroll
        for (int v = 0; v < 8; ++v) rowsum += rmin[r][v];
    }
    rowsum += __shfl_xor(rowsum, 16, 32);    // combine halves: rows 0-7 + rows 8-15

    if (lane == 0)
        __hip_atomic_fetch_add(out, rowsum * scale,
                               __ATOMIC_RELAXED, __HIP_MEMORY_SCOPE_AGENT);
}

extern "C" void kernel_launch(void* const* d_in, const int* in_sizes, int n_in,
                              void* d_out, int out_size, void* d_ws, size_t ws_size,
                              hipStream_t stream) {
    const float* x = (const float*)d_in[0];   // [B, N, 3]
    const float* y = (const float*)d_in[1];   // [B, M, 3]
    float* out = (float*)d_out;               // scalar

    const int B = 4;
    const int N = in_sizes[0] / (B * 3);
    const int M = in_sizes[1] / (B * 3);

    chamfer_zero_kernel<<<1, 1, 0, stream>>>(out);

    // Pass 1: for each x, min over y.   contribution scale = 1/(B*N)
    {
        const int pairs  = B * (N >> 4) / ROWTILES;
        const int blocks = (pairs + WAVES_PER_BLOCK - 1) / WAVES_PER_BLOCK;
        chamfer_dir_kernel<<<blocks, BLOCK_THREADS, 0, stream>>>(
            x, y, N, M, pairs, 1.0f / (float)(B * N), out);
    }
    // Pass 2: for each y, min over x.   contribution scale = 1/(B*M)
    {
        const int pairs  = B * (M >> 4) / ROWTILES;
        const int blocks = (pairs + WAVES_PER_BLOCK - 1) / WAVES_PER_BLOCK;
        chamfer_dir_kernel<<<blocks, BLOCK_THREADS, 0, stream>>>(
            y, x, M, N, pairs, 1.0f / (float)(B * M), out);
    }
}